// MicroCoupledSuperNet_16784732192989
// MI455X (gfx1250) — compile-verified
//
#include <hip/hip_runtime.h>
#include <hip/hip_bf16.h>
#include <math.h>

#define NN 50000
#define EE 640000
#define D_IN 128
#define HH 128
#define D_OUT 64
#define GG 500
#define LL 2
#define EPS 1e-5f

typedef __attribute__((ext_vector_type(2))) float v2f;
typedef __attribute__((ext_vector_type(8))) float v8f;

// ---------------------------------------------------------------------------
// WMMA GEMM: C[rows x ncols] = A[rows x 128] (optionally row-scaled) @ W[128 x ncols]
//            (+ bias) (+ Cin accumulate). One block = one 16-row tile,
//            one wave per 16-column tile (blockDim.x = 32 * ncols/16).
// Uses V_WMMA_F32_16X16X4_F32 (fp32 in, fp32 acc) -- exact fp32 math.
// SCALE/ACC are compile-time so the inner loop and epilogue stay branch-free.
// ---------------------------------------------------------------------------
template <bool SCALE, bool ACC>
__global__ void gemm_wmma(const float* __restrict__ A,
                          const float* __restrict__ rowscale,   // used iff SCALE
                          const float* __restrict__ W,          // [128 x ncols]
                          const float* __restrict__ bias,       // or nullptr
                          const float* __restrict__ Cin,        // used iff ACC
                          float* __restrict__ Cout,
                          int nrows, int ncols, int rowlimit)
{
    const int KDIM = 128;
    __shared__ float As[16][132];            // padded: 16 rows stride 132 -> no bank conflicts

    const int rowbase = blockIdx.x * 16;
    const int tid  = threadIdx.x;
    const int wave = tid >> 5;
    const int lane = tid & 31;

    // cooperative A-tile load (16 x 128 floats as float4)
    for (int i = tid; i < 16 * 32; i += blockDim.x) {
        int r  = i >> 5;
        int c4 = (i & 31) * 4;
        int row = rowbase + r;
        float4 v = make_float4(0.f, 0.f, 0.f, 0.f);
        if (row < nrows) v = *(const float4*)(A + (size_t)row * KDIM + c4);
        As[r][c4 + 0] = v.x; As[r][c4 + 1] = v.y;
        As[r][c4 + 2] = v.z; As[r][c4 + 3] = v.w;
    }
    __syncthreads();

    const int m     = lane & 15;   // A row within tile (both half-waves)
    const int khalf = lane >> 4;   // 0: K=k0,k0+1   1: K=k0+2,k0+3
    const int n     = wave * 16 + (lane & 15);

    float rs = 1.0f;
    if (SCALE) {
        int row = rowbase + m;
        rs = (row < nrows) ? rowscale[row] : 0.0f;
    }

    v8f acc = {};
    for (int k0 = 0; k0 < KDIM; k0 += 4) {
        const int k = k0 + khalf * 2;
        v2f a, b;
        a.x = As[m][k];
        a.y = As[m][k + 1];
        if (SCALE) { a.x *= rs; a.y *= rs; }
        b.x = W[(size_t)k       * ncols + n];
        b.y = W[(size_t)(k + 1) * ncols + n];
        acc = __builtin_amdgcn_wmma_f32_16x16x4_f32(
                  false, a, false, b, (short)0, acc, false, false);
    }

    const float bv = bias ? bias[n] : 0.0f;
    const int rmax = (nrows < rowlimit) ? nrows : rowlimit;
    if (rowbase + 16 <= rmax) {
        // full tile: branch-free epilogue (all node GEMMs, since 50000 % 16 == 0)
        #pragma unroll
        for (int v = 0; v < 8; ++v) {
            int row = rowbase + v + 8 * khalf;  // D: vgpr v -> M=v (lanes0-15), M=8+v (lanes16-31)
            size_t idx = (size_t)row * ncols + n;
            float val = acc[v] + bv;
            if (ACC) val += Cin[idx];
            Cout[idx] = val;
        }
    } else {
        #pragma unroll
        for (int v = 0; v < 8; ++v) {
            int row = rowbase + v + 8 * khalf;
            if (row < rmax) {
                size_t idx = (size_t)row * ncols + n;
                float val = acc[v] + bv;
                if (ACC) val += Cin[idx];
                Cout[idx] = val;
            }
        }
    }
}

// ---------------------------------------------------------------------------
// Degree count: cnt[dst] += 1 per edge
// ---------------------------------------------------------------------------
__global__ void deg_kernel(const int* __restrict__ ei, float* __restrict__ cnt)
{
    int e = blockIdx.x * blockDim.x + threadIdx.x;
    if (e < EE) atomicAdd(cnt + ei[EE + e], 1.0f);
}

// dinv[n] = rsqrt(deg_with_selfloop); invdegn[n] = 1/max(deg,1)
__global__ void prep_kernel(const float* __restrict__ cnt,
                            float* __restrict__ dinv, float* __restrict__ invdegn)
{
    int nidx = blockIdx.x * blockDim.x + threadIdx.x;
    if (nidx < NN) {
        float c = cnt[nidx];
        dinv[nidx]    = rsqrtf(c + 1.0f);
        invdegn[nidx] = 1.0f / fmaxf(c, 1.0f);
    }
}

// ---------------------------------------------------------------------------
// Fused edge scatter: 32 lanes per edge, float4 features.
//   gcnagg[dst] += hw[src] * dinv[src]*dinv[dst]
//   meanacc[dst] += h[src]
// ---------------------------------------------------------------------------
__global__ void edge_scatter(const int* __restrict__ ei,
                             const float* __restrict__ dinv,
                             const float* __restrict__ hw,
                             const float* __restrict__ h,
                             float* __restrict__ gcnagg,
                             float* __restrict__ meanacc)
{
    size_t tid = (size_t)blockIdx.x * blockDim.x + threadIdx.x;
    size_t e = tid >> 5;
    if (e >= EE) return;
    int f = (int)(tid & 31) * 4;
    int s = ei[e];
    int d = ei[EE + e];
    float nr = dinv[s] * dinv[d];
    float4 hv = *(const float4*)(hw + (size_t)s * HH + f);
    float4 xv = *(const float4*)(h  + (size_t)s * HH + f);
    float* gp = gcnagg  + (size_t)d * HH + f;
    float* mp = meanacc + (size_t)d * HH + f;
    atomicAdd(gp + 0, hv.x * nr); atomicAdd(gp + 1, hv.y * nr);
    atomicAdd(gp + 2, hv.z * nr); atomicAdd(gp + 3, hv.w * nr);
    atomicAdd(mp + 0, xv.x); atomicAdd(mp + 1, xv.y);
    atomicAdd(mp + 2, xv.z); atomicAdd(mp + 3, xv.w);
}

// ---------------------------------------------------------------------------
// Combine: one wave per node (4 features per lane).
//   gcn  = gcnagg + hw*dinv^2 (self-loop) + gcn_b
//   hnew = wc0*gcn + wc1*sage
//   h2   = wn0*layernorm(hnew)*g+b + wn1*hnew
//   out  = wa0*relu + wa1*tanh + wa2*elu
//   h <- out ; skip <- out (first) or skip+out
// ---------------------------------------------------------------------------
__global__ void combine_kernel(const float* __restrict__ gcnagg,
                               const float* __restrict__ hwbuf,
                               const float* __restrict__ sage,
                               const float* __restrict__ dinv,
                               const float* __restrict__ gcn_b,
                               const float* __restrict__ ln_g,
                               const float* __restrict__ ln_b,
                               const float* __restrict__ a_conv,
                               const float* __restrict__ a_norm,
                               const float* __restrict__ a_act,
                               int layer,
                               float* __restrict__ hout,
                               float* __restrict__ skip,
                               int first)
{
    int node = blockIdx.x * (blockDim.x >> 5) + (threadIdx.x >> 5);
    if (node >= NN) return;
    int lane = threadIdx.x & 31;
    int f = lane * 4;

    // softmax architecture weights (TEMP = 1)
    float c0 = a_conv[layer * 2 + 0], c1 = a_conv[layer * 2 + 1];
    float mx = fmaxf(c0, c1);
    float e0 = __expf(c0 - mx), e1 = __expf(c1 - mx);
    float wc0 = e0 / (e0 + e1), wc1 = e1 / (e0 + e1);

    float n0 = a_norm[layer * 2 + 0], n1 = a_norm[layer * 2 + 1];
    mx = fmaxf(n0, n1);
    float p0 = __expf(n0 - mx), p1 = __expf(n1 - mx);
    float wn0 = p0 / (p0 + p1), wn1 = p1 / (p0 + p1);

    float a0 = a_act[layer * 3 + 0], a1 = a_act[layer * 3 + 1], a2 = a_act[layer * 3 + 2];
    mx = fmaxf(fmaxf(a0, a1), a2);
    float g0 = __expf(a0 - mx), g1 = __expf(a1 - mx), g2 = __expf(a2 - mx);
    float gs = g0 + g1 + g2;
    float wa0 = g0 / gs, wa1 = g1 / gs, wa2 = g2 / gs;

    size_t base = (size_t)node * HH + f;
    const float* aggp = gcnagg + base;
    const float* hwp  = hwbuf + base;
    const float* sgp  = sage + base;
    const float* gbp  = gcn_b + (size_t)layer * HH + f;
    float dv = dinv[node];
    float dv2 = dv * dv;

    float hn[4];
    #pragma unroll
    for (int i = 0; i < 4; ++i) {
        float gcn = aggp[i] + hwp[i] * dv2 + gbp[i];
        hn[i] = wc0 * gcn + wc1 * sgp[i];
    }

    // layernorm over H=128 via wave32 reduction
    float s = hn[0] + hn[1] + hn[2] + hn[3];
    for (int off = 16; off; off >>= 1) s += __shfl_xor(s, off, 32);
    float mu = s * (1.0f / HH);
    float vs = 0.f;
    #pragma unroll
    for (int i = 0; i < 4; ++i) { float d = hn[i] - mu; vs += d * d; }
    for (int off = 16; off; off >>= 1) vs += __shfl_xor(vs, off, 32);
    float inv = rsqrtf(vs * (1.0f / HH) + EPS);

    const float* lgp = ln_g + (size_t)layer * HH + f;
    const float* lbp = ln_b + (size_t)layer * HH + f;
    float* op = hout + base;
    float* sp = skip + base;
    #pragma unroll
    for (int i = 0; i < 4; ++i) {
        float ln = (hn[i] - mu) * inv * lgp[i] + lbp[i];
        float h2 = wn0 * ln + wn1 * hn[i];
        float r  = fmaxf(h2, 0.f);
        float t  = tanhf(h2);
        float el = (h2 > 0.f) ? h2 : (__expf(h2) - 1.0f);
        float out = wa0 * r + wa1 * t + wa2 * el;
        op[i] = out;
        sp[i] = first ? out : (sp[i] + out);
    }
}

// ---------------------------------------------------------------------------
// Sum pooling: pooled[batch[node]] += skip[node]
// ---------------------------------------------------------------------------
__global__ void pool_kernel(const float* __restrict__ skip,
                            const int* __restrict__ batch,
                            float* __restrict__ pooled)
{
    size_t tid = (size_t)blockIdx.x * blockDim.x + threadIdx.x;
    size_t node = tid >> 5;
    if (node >= NN) return;
    int f = (int)(tid & 31) * 4;
    int g = batch[node];
    const float* spv = skip + node * HH + f;
    float* pp = pooled + (size_t)g * HH + f;
    atomicAdd(pp + 0, spv[0]); atomicAdd(pp + 1, spv[1]);
    atomicAdd(pp + 2, spv[2]); atomicAdd(pp + 3, spv[3]);
}

// ---------------------------------------------------------------------------
extern "C" void kernel_launch(void* const* d_in, const int* in_sizes, int n_in,
                              void* d_out, int out_size, void* d_ws, size_t ws_size,
                              hipStream_t stream)
{
    const float* x      = (const float*)d_in[0];
    const int*   ei     = (const int*)d_in[1];
    const int*   batch  = (const int*)d_in[2];
    const float* pre_w  = (const float*)d_in[3];
    const float* pre_b  = (const float*)d_in[4];
    const float* post_w = (const float*)d_in[5];
    const float* post_b = (const float*)d_in[6];
    const float* gcn_w  = (const float*)d_in[7];
    const float* gcn_b  = (const float*)d_in[8];
    const float* sage_ws = (const float*)d_in[9];
    const float* sage_wn = (const float*)d_in[10];
    const float* ln_g   = (const float*)d_in[11];
    const float* ln_b   = (const float*)d_in[12];
    const float* a_conv = (const float*)d_in[13];
    const float* a_norm = (const float*)d_in[14];
    const float* a_act  = (const float*)d_in[15];
    float* out = (float*)d_out;

    char* ws = (char*)d_ws;
    const size_t NH = (size_t)NN * HH * sizeof(float);
    float* h      = (float*)(ws);             ws += NH;
    float* hw     = (float*)(ws);             ws += NH;
    float* gcnagg = (float*)(ws);             ws += NH;
    float* meanb  = (float*)(ws);             ws += NH;
    float* sage   = (float*)(ws);             ws += NH;
    float* skip   = (float*)(ws);             ws += NH;
    float* cnt    = (float*)(ws);             ws += (size_t)NN * sizeof(float);
    float* dinv   = (float*)(ws);             ws += (size_t)NN * sizeof(float);
    float* invdgn = (float*)(ws);             ws += (size_t)NN * sizeof(float);
    float* pooled = (float*)(ws);             ws += (size_t)GG * HH * sizeof(float);

    const int mtilesN = NN / 16 + ((NN % 16) ? 1 : 0);   // 3125
    const int mtilesG = GG / 16 + ((GG % 16) ? 1 : 0);   // 32
    const int edgeBlocks = (int)(((size_t)EE * 32 + 255) / 256);
    const int nodeBlocks = (int)(((size_t)NN * 32 + 255) / 256);

    // degrees
    hipMemsetAsync(cnt, 0, (size_t)NN * sizeof(float), stream);
    deg_kernel<<<(EE + 255) / 256, 256, 0, stream>>>(ei, cnt);
    prep_kernel<<<(NN + 255) / 256, 256, 0, stream>>>(cnt, dinv, invdgn);

    // pre-process MLP: h = x @ pre_w + pre_b
    gemm_wmma<false, false><<<mtilesN, 256, 0, stream>>>(
        x, nullptr, pre_w, pre_b, nullptr, h, NN, HH, NN);

    for (int l = 0; l < LL; ++l) {
        const float* wg  = gcn_w  + (size_t)l * HH * HH;
        const float* wss = sage_ws + (size_t)l * HH * HH;
        const float* wsn = sage_wn + (size_t)l * HH * HH;

        // hw = h @ gcn_w[l]
        gemm_wmma<false, false><<<mtilesN, 256, 0, stream>>>(
            h, nullptr, wg, nullptr, nullptr, hw, NN, HH, NN);

        // scatter: gcnagg += norm*hw[src]; mean += h[src]
        hipMemsetAsync(gcnagg, 0, NH, stream);
        hipMemsetAsync(meanb, 0, NH, stream);
        edge_scatter<<<edgeBlocks, 256, 0, stream>>>(ei, dinv, hw, h, gcnagg, meanb);

        // sage = h @ sage_ws[l] + (mean/deg_n) @ sage_wn[l]
        gemm_wmma<false, false><<<mtilesN, 256, 0, stream>>>(
            h, nullptr, wss, nullptr, nullptr, sage, NN, HH, NN);
        gemm_wmma<true, true><<<mtilesN, 256, 0, stream>>>(
            meanb, invdgn, wsn, nullptr, sage, sage, NN, HH, NN);

        // combine conv mix + norm mix + act mix, update h and skip
        combine_kernel<<<(NN + 7) / 8, 256, 0, stream>>>(
            gcnagg, hw, sage, dinv, gcn_b, ln_g, ln_b,
            a_conv, a_norm, a_act, l, h, skip, (l == 0) ? 1 : 0);
    }

    // sum pooling + post MLP
    hipMemsetAsync(pooled, 0, (size_t)GG * HH * sizeof(float), stream);
    pool_kernel<<<nodeBlocks, 256, 0, stream>>>(skip, batch, pooled);
    gemm_wmma<false, false><<<mtilesG, 128, 0, stream>>>(
        pooled, nullptr, post_w, post_b, nullptr, out, GG, D_OUT, GG);
}